// SimpleDialogGNN_7859790152085
// MI455X (gfx1250) — compile-verified
//
#include <hip/hip_runtime.h>
#include <hip/hip_bf16.h>

typedef float v2f __attribute__((ext_vector_type(2)));
typedef float v8f __attribute__((ext_vector_type(8)));

#define B_  8
#define T_  1024
#define H_  1024
#define M_  (B_ * T_)
#define WIN 17
#define LN_EPS 1e-5f

// ---------------------------------------------------------------------------
// Kernel 1: windowed speaker-weighted message aggregation
// one block per (b,t); 256 threads each produce 4 of the H=1024 outputs
// ---------------------------------------------------------------------------
__global__ __launch_bounds__(256) void msg_kernel(
    const float* __restrict__ x, const float* __restrict__ qmask,
    const int* __restrict__ dia_len, float* __restrict__ msg)
{
    int bt = blockIdx.x;            // 0 .. B*T-1
    int b  = bt >> 10;              // T = 1024
    int t  = bt & (T_ - 1);

    __shared__ float sw[WIN];
    __shared__ int   sidx[WIN];
    __shared__ float sinv;

    if (threadIdx.x == 0) {
        int   L  = dia_len[b];
        float q0 = qmask[(size_t)bt * 2 + 0];
        float q1 = qmask[(size_t)bt * 2 + 1];
        int spk_c = (q1 > q0) ? 1 : 0;
        int cnt = 0;
        for (int i = 0; i < WIN; ++i) {
            int  idx   = t - 8 + i;
            bool valid = (idx >= 0) && (idx <= L - 1);
            int  idxc  = min(max(idx, 0), T_ - 1);
            sidx[i] = idxc;
            float n0 = qmask[((size_t)b * T_ + idxc) * 2 + 0];
            float n1 = qmask[((size_t)b * T_ + idxc) * 2 + 1];
            int spk_n = (n1 > n0) ? 1 : 0;
            sw[i] = valid ? ((spk_n == spk_c) ? 1.0f : 0.5f) : 0.0f;
            cnt  += valid ? 1 : 0;
        }
        sinv = 1.0f / (float)max(cnt, 1);
    }
    __syncthreads();

    int h0 = threadIdx.x * 4;
    float4 acc = make_float4(0.f, 0.f, 0.f, 0.f);
#pragma unroll
    for (int i = 0; i < WIN; ++i) {
        float w = sw[i];
        const float4 v = *(const float4*)&x[((size_t)b * T_ + sidx[i]) * H_ + h0];
        acc.x += w * v.x; acc.y += w * v.y; acc.z += w * v.z; acc.w += w * v.w;
    }
    float s = sinv;
    float4 o = make_float4(acc.x * s, acc.y * s, acc.z * s, acc.w * s);
    *(float4*)&msg[(size_t)bt * H_ + h0] = o;
}

// ---------------------------------------------------------------------------
// Kernel 2/3: WMMA FP32 GEMM  C = A0@W0 (+ A1@W1) + epilogue
//   workgroup tile 64(M) x 128(N), 8 waves (2x4), each wave 32x32 = 2x2 WMMA
//   K staged through LDS in 16-wide slabs, v_wmma_f32_16x16x4_f32 inner loop.
//   A tile: row-major, stride 20 (bank-conflict-free, b64-aligned K pairs).
//   W tile: K-pair interleaved -> lBp[kp][col] = {W[2kp][col], W[2kp+1][col]}
//           so each B-fragment is a single ds_load_b64 (no register packing);
//           row stride 288 floats puts the kp / kp+1 lane-halves on disjoint
//           bank halves.
// mode 0: out = relu( (t < dia_len[b]) ? C + bias0[n] + bias1[n] : x[m,n] )
// mode 1: out = x[m,n] + C + bias0[n]
// ---------------------------------------------------------------------------
__global__ __launch_bounds__(256) void gemm_wmma(
    const float* __restrict__ A0, const float* __restrict__ W0,
    const float* __restrict__ A1, const float* __restrict__ W1,
    const float* __restrict__ bias0, const float* __restrict__ bias1,
    const float* __restrict__ xres, const int* __restrict__ dia_len,
    float* __restrict__ out, int mode)
{
    const int N = H_, K = H_;
    __shared__ float lA[64][20];     // 5120 B
    __shared__ float lBp[8][288];    // 9216 B ; pairs at [kp][2*col]

    int tid   = threadIdx.x;
    int wave  = tid >> 5;
    int lane  = tid & 31;
    int waveM = wave >> 2;           // 0..1
    int waveN = wave & 3;            // 0..3
    int mBase = blockIdx.x * 64;
    int nBase = blockIdx.y * 128;

    v8f acc[2][2] = {};

    int npairs = (A1 != nullptr) ? 2 : 1;
    for (int p = 0; p < npairs; ++p) {
        const float* __restrict__ A = p ? A1 : A0;
        const float* __restrict__ W = p ? W1 : W0;

        for (int kb = 0; kb < K; kb += 16) {
            // stage A tile 64x16 (one float4 per thread)
            {
                int tRow = tid >> 2;
                int tCol = (tid & 3) * 4;
                const float4 v =
                    *(const float4*)&A[(size_t)(mBase + tRow) * K + kb + tCol];
                lA[tRow][tCol + 0] = v.x;
                lA[tRow][tCol + 1] = v.y;
                lA[tRow][tCol + 2] = v.z;
                lA[tRow][tCol + 3] = v.w;
            }
            // stage W tile 16x128, K-pair interleaved (two float4 per thread)
            {
                int kp = tid >> 5;            // 0..7  (K rows 2kp, 2kp+1)
                int c  = (tid & 31) * 4;      // 0..124
                const float4 r0 =
                    *(const float4*)&W[(size_t)(kb + 2 * kp) * N + nBase + c];
                const float4 r1 =
                    *(const float4*)&W[(size_t)(kb + 2 * kp + 1) * N + nBase + c];
                float4 p0 = make_float4(r0.x, r1.x, r0.y, r1.y);
                float4 p1 = make_float4(r0.z, r1.z, r0.w, r1.w);
                *(float4*)&lBp[kp][c * 2]     = p0;
                *(float4*)&lBp[kp][c * 2 + 4] = p1;
            }
            // prefetch next K slab into cache (global_prefetch_b8)
            if (kb + 16 < K) {
                int tRow = tid >> 2;
                int tCol = (tid & 3) * 4;
                __builtin_prefetch(&A[(size_t)(mBase + tRow) * K + kb + 16 + tCol], 0, 3);
                int kp = tid >> 5;
                int c  = (tid & 31) * 4;
                __builtin_prefetch(&W[(size_t)(kb + 16 + 2 * kp) * N + nBase + c], 0, 3);
            }
            __syncthreads();

            int lsub  = lane & 15;
            int khalf = (lane >> 4) * 2;      // lanes 0-15: K=0,1 ; 16-31: K=2,3
#pragma unroll
            for (int ks = 0; ks < 4; ++ks) {
                int k   = ks * 4;
                int kp2 = (k + khalf) >> 1;   // interleaved pair row
                v2f afrag[2], bfrag[2];
#pragma unroll
                for (int i = 0; i < 2; ++i) {
                    int row = waveM * 32 + i * 16 + lsub;
                    afrag[i] = *(const v2f*)&lA[row][k + khalf];
                }
#pragma unroll
                for (int j = 0; j < 2; ++j) {
                    int col = waveN * 32 + j * 16 + lsub;
                    bfrag[j] = *(const v2f*)&lBp[kp2][col * 2];
                }
#pragma unroll
                for (int i = 0; i < 2; ++i)
#pragma unroll
                    for (int j = 0; j < 2; ++j)
                        acc[i][j] = __builtin_amdgcn_wmma_f32_16x16x4_f32(
                            false, afrag[i], false, bfrag[j],
                            (short)0, acc[i][j], false, false);
            }
            __syncthreads();
        }
    }

    // epilogue: C/D layout -> VGPR r: lanes 0-15 = row r, lanes 16-31 = row r+8
    int lsub = lane & 15;
    int hi   = (lane >> 4) * 8;
#pragma unroll
    for (int i = 0; i < 2; ++i) {
#pragma unroll
        for (int j = 0; j < 2; ++j) {
            int n = nBase + waveN * 32 + j * 16 + lsub;
#pragma unroll
            for (int r = 0; r < 8; ++r) {
                int    m   = mBase + waveM * 32 + i * 16 + r + hi;
                float  c   = acc[i][j][r];
                size_t off = (size_t)m * N + n;
                float  val;
                if (mode == 0) {
                    int  bb    = m >> 10;
                    int  tt    = m & (T_ - 1);
                    bool valid = tt < dia_len[bb];
                    float v    = valid ? (c + bias0[n] + bias1[n]) : xres[off];
                    val = fmaxf(v, 0.0f);
                } else {
                    val = xres[off] + c + bias0[n];
                }
                out[off] = val;
            }
        }
    }
}

// ---------------------------------------------------------------------------
// Kernel 4: LayerNorm over H=1024, one block (8 wave32) per row
// ---------------------------------------------------------------------------
__global__ __launch_bounds__(256) void ln_kernel(
    const float* __restrict__ z, const float* __restrict__ gamma,
    const float* __restrict__ beta, float* __restrict__ out)
{
    int row  = blockIdx.x;
    int tid  = threadIdx.x;
    int wave = tid >> 5;
    int lane = tid & 31;

    const float4 v = *(const float4*)&z[(size_t)row * H_ + tid * 4];
    float s1 = v.x + v.y + v.z + v.w;
    float s2 = v.x * v.x + v.y * v.y + v.z * v.z + v.w * v.w;

#pragma unroll
    for (int off = 16; off > 0; off >>= 1) {
        s1 += __shfl_xor(s1, off, 32);
        s2 += __shfl_xor(s2, off, 32);
    }
    __shared__ float ws1[8], ws2[8];
    if (lane == 0) { ws1[wave] = s1; ws2[wave] = s2; }
    __syncthreads();
    if (tid == 0) {
        float a = 0.f, b = 0.f;
#pragma unroll
        for (int i = 0; i < 8; ++i) { a += ws1[i]; b += ws2[i]; }
        float mean = a * (1.0f / H_);
        float var  = b * (1.0f / H_) - mean * mean;
        ws1[0] = mean;
        ws2[0] = __frsqrt_rn(var + LN_EPS);
    }
    __syncthreads();
    float mean = ws1[0];
    float rstd = ws2[0];

    int h0 = tid * 4;
    const float4 g  = *(const float4*)&gamma[h0];
    const float4 be = *(const float4*)&beta[h0];
    float4 o;
    o.x = g.x * (v.x - mean) * rstd + be.x;
    o.y = g.y * (v.y - mean) * rstd + be.y;
    o.z = g.z * (v.z - mean) * rstd + be.z;
    o.w = g.w * (v.w - mean) * rstd + be.w;
    *(float4*)&out[(size_t)row * H_ + h0] = o;
}

// ---------------------------------------------------------------------------
extern "C" void kernel_launch(void* const* d_in, const int* in_sizes, int n_in,
                              void* d_out, int out_size, void* d_ws, size_t ws_size,
                              hipStream_t stream)
{
    const float* x       = (const float*)d_in[0];
    const float* qmask   = (const float*)d_in[1];
    const int*   dia_len = (const int*)d_in[2];
    // d_in[3] = wp (=8), d_in[4] = wf (=8): fixed by problem size, window = 17
    const float* W_msg   = (const float*)d_in[5];
    const float* b_msg   = (const float*)d_in[6];
    const float* W_self  = (const float*)d_in[7];
    const float* b_self  = (const float*)d_in[8];
    const float* W_out   = (const float*)d_in[9];
    const float* b_out   = (const float*)d_in[10];
    const float* gamma   = (const float*)d_in[11];
    const float* beta    = (const float*)d_in[12];

    float* msg = (float*)d_ws;                        // 8192*1024 f32 = 32 MB
    float* a2  = msg + (size_t)M_ * H_;               // next 32 MB
    float* zb  = msg;                                 // msg dead after gemm0

    // 1) message aggregation
    msg_kernel<<<M_, 256, 0, stream>>>(x, qmask, dia_len, msg);

    // 2) h = x@W_self + msg@W_msg + biases ; valid-select ; relu -> a2
    dim3 g2(M_ / 64, H_ / 128);
    gemm_wmma<<<g2, 256, 0, stream>>>(x, W_self, msg, W_msg,
                                      b_self, b_msg, x, dia_len, a2, 0);

    // 3) z = x + a2@W_out + b_out -> zb
    gemm_wmma<<<g2, 256, 0, stream>>>(a2, W_out, nullptr, nullptr,
                                      b_out, nullptr, x, dia_len, zb, 1);

    // 4) LayerNorm -> d_out
    ln_kernel<<<M_, 256, 0, stream>>>(zb, gamma, beta, (float*)d_out);
}